// BiRWKVLayer_56624848830922
// MI455X (gfx1250) — compile-verified
//
#include <hip/hip_runtime.h>
#include <stdint.h>

// ---------------------------------------------------------------------------
// BiRWKV layer for MI455X (gfx1250):
//   GEMM1 (fp32 WMMA 16x16x4) -> bidirectional WKV scan (wave-parallel over
//   B*2*C channels, fused sigmoid(r)*y written in-place into the r slots of
//   rkv) -> GEMM2 (fp32 WMMA, split-K column remap reads straight from rkv).
// TDM (tensor_load_to_lds) double-buffers LDS panels for both GEMMs.
// ---------------------------------------------------------------------------

typedef float v2f  __attribute__((ext_vector_type(2)));
typedef float v8f  __attribute__((ext_vector_type(8)));
typedef unsigned int u32x4 __attribute__((ext_vector_type(4)));
typedef int   i32x4 __attribute__((ext_vector_type(4)));
typedef int   i32x8 __attribute__((ext_vector_type(8)));

#if defined(__has_builtin)
#  if __has_builtin(__builtin_amdgcn_tensor_load_to_lds)
#    define HAVE_TDM 1
#  endif
#endif
#ifndef HAVE_TDM
#  define HAVE_TDM 0
#endif

#define BM 64
#define BN 256
#define KC 16
#define NTHREADS 256

#if HAVE_TDM
// Build a 2D Tensor-DMA descriptor (D#) and issue TENSOR_LOAD_TO_LDS.
// tile0 = contiguous elements per row, tile1 = rows, stride0 = row stride
// (elements, fp32 => data_size code 2). tensor_dim == tile_dim (in-bounds).
__device__ __forceinline__ void tdm_load_2d(unsigned lds_off_bytes,
                                            const float* gptr,
                                            unsigned tile0, unsigned tile1,
                                            unsigned long long stride0) {
  unsigned long long ga = (unsigned long long)(uintptr_t)gptr;
  u32x4 g0;
  g0[0] = 1u;                                   // count=1, user descriptor
  g0[1] = lds_off_bytes;                        // D#.lds_addr
  g0[2] = (unsigned)(ga & 0xFFFFFFFFu);         // global_addr[31:0]
  g0[3] = (unsigned)((ga >> 32) & 0x01FFFFFFu)  // global_addr[56:32]
          | 0x80000000u;                        // type=2 ("image")
  i32x8 g1;
  g1[0] = 0x00020000;                           // data_size=2 (4B), mask=0
  g1[1] = (int)((tile0 & 0xFFFFu) << 16);       // tensor_dim0[15:0]
  g1[2] = (int)((tile0 >> 16) | ((tile1 & 0xFFFFu) << 16)); // dim0 hi | dim1 lo
  g1[3] = (int)((tile1 >> 16) | (tile0 << 16)); // dim1 hi | tile_dim0
  g1[4] = (int)(tile1 & 0xFFFFu);               // tile_dim1 (tile_dim2=0)
  g1[5] = (int)(unsigned)(stride0 & 0xFFFFFFFFull);      // dim0_stride[31:0]
  g1[6] = (int)(unsigned)((stride0 >> 32) & 0xFFFFull);  // dim0_stride[47:32]
  g1[7] = 0;
  i32x4 z4 = {0, 0, 0, 0};
#if __clang_major__ >= 23
  i32x8 z8 = {0, 0, 0, 0, 0, 0, 0, 0};
  __builtin_amdgcn_tensor_load_to_lds(g0, g1, z4, z4, z8, 0);
#else
  __builtin_amdgcn_tensor_load_to_lds(g0, g1, z4, z4, 0);
#endif
}
#endif  // HAVE_TDM

// The TDM writes LDS behind the compiler's back (the LDS address travels
// inside the D# as an integer). Escape the LDS pointer through an asm with a
// memory clobber so LLVM treats the panel as written — otherwise the ds_load
// fragment reads get folded to undef and deleted (observed in round 1).
__device__ __forceinline__ void lds_publish(const void* p) {
  asm volatile("" : : "r"(p) : "memory");
}

// ---------------------------------------------------------------------------
// fp32 WMMA GEMM: C[M,N] = Aeff[M,K] * B[K,N]
//   Aeff(m,kk) = A[m*lda + kk + (kk >= kSplit ? kSkip : 0)]
// Block: 256 threads (8 waves), 64x256 tile; wave = 2x4 WMMA 16x16 tiles.
// LDS double-buffered A(64x16)/B(16x256) panels staged by the TDM.
// ---------------------------------------------------------------------------
__global__ __launch_bounds__(NTHREADS)
void wmma_gemm_f32(const float* __restrict__ A, const float* __restrict__ Bm,
                   float* __restrict__ Cm, int M, int N, int K,
                   int lda, int kSplit, int kSkip) {
  constexpr int PANEL = BM * KC + KC * BN;      // floats per buffer
  __shared__ float smem[2 * PANEL];             // 40 KB (of 320 KB/WGP)

  const int tid  = threadIdx.x;
  const int lane = tid & 31;
  const int wave = tid >> 5;
  const int hsel = lane >> 4;                   // lane half (K/M split)
  const int lm   = lane & 15;
  const int wm   = wave & 1;                    // 2 waves along M
  const int wn   = wave >> 1;                   // 4 waves along N
  const int mBlk = blockIdx.y * BM;
  const int nBlk = blockIdx.x * BN;
  const int nk   = K / KC;

  v8f acc[2][4] = {};

#if HAVE_TDM
  if (tid < 32) {  // one wave issues DMA for the whole workgroup
    const int add0 = (0 >= kSplit) ? kSkip : 0;
    tdm_load_2d(0u, A + (size_t)mBlk * lda + add0, KC, BM,
                (unsigned long long)lda);
    tdm_load_2d((unsigned)(BM * KC) * 4u, Bm + (size_t)nBlk, BN, KC,
                (unsigned long long)N);
  }
#endif

  for (int kc = 0; kc < nk; ++kc) {
    const int cur = kc & 1;
    float* sA = smem + cur * PANEL;
    float* sB = sA + BM * KC;

#if HAVE_TDM
    __builtin_amdgcn_s_wait_tensorcnt(0);       // DMA for chunk kc done
    __syncthreads();                            // publish LDS; prev compute done
    lds_publish(smem);                          // make TDM's LDS writes visible
    if (tid < 32 && (kc + 1) < nk) {            // prefetch chunk kc+1
      const int nxt  = (kc + 1) & 1;
      const int kk0  = (kc + 1) * KC;
      const int addA = (kk0 >= kSplit) ? kSkip : 0;
      const unsigned offA = (unsigned)(nxt * PANEL) * 4u;
      tdm_load_2d(offA, A + (size_t)mBlk * lda + kk0 + addA, KC, BM,
                  (unsigned long long)lda);
      tdm_load_2d(offA + (unsigned)(BM * KC) * 4u,
                  Bm + (size_t)kk0 * N + nBlk, BN, KC,
                  (unsigned long long)N);
    }
#else
    __syncthreads();
    {  // cooperative staging fallback
      const int kk0  = kc * KC;
      const int addA = (kk0 >= kSplit) ? kSkip : 0;
      for (int i = tid; i < BM * KC; i += NTHREADS) {
        int r = i / KC, cc = i - r * KC;
        sA[i] = A[(size_t)(mBlk + r) * lda + kk0 + addA + cc];
      }
      for (int i = tid; i < KC * BN; i += NTHREADS) {
        int r = i / BN, cc = i - r * BN;
        sB[i] = Bm[(size_t)(kk0 + r) * N + nBlk + cc];
      }
    }
    __syncthreads();
#endif

    // 4 WMMA K-steps of 4 per 16-wide chunk
    #pragma unroll
    for (int ks = 0; ks < KC / 4; ++ks) {
      const int k0 = ks * 4 + hsel * 2;
      v2f afr[2];
      #pragma unroll
      for (int mt = 0; mt < 2; ++mt) {
        const int mrow = wm * 32 + mt * 16 + lm;
        afr[mt] = *(const v2f*)&sA[mrow * KC + k0];   // A 16x4 frag (2 VGPR)
      }
      v2f bfr[4];
      #pragma unroll
      for (int nt = 0; nt < 4; ++nt) {
        const int ncol = wn * 64 + nt * 16 + lm;
        v2f b;
        b.x = sB[(k0 + 0) * BN + ncol];               // B 4x16 frag
        b.y = sB[(k0 + 1) * BN + ncol];
        bfr[nt] = b;
      }
      #pragma unroll
      for (int mt = 0; mt < 2; ++mt)
        #pragma unroll
        for (int nt = 0; nt < 4; ++nt)
          acc[mt][nt] = __builtin_amdgcn_wmma_f32_16x16x4_f32(
              false, afr[mt], false, bfr[nt], (short)0, acc[mt][nt],
              false, false);
    }
  }

  // Epilogue: C/D layout — VGPR i: lanes 0-15 -> M=i, lanes 16-31 -> M=i+8
  #pragma unroll
  for (int mt = 0; mt < 2; ++mt) {
    #pragma unroll
    for (int nt = 0; nt < 4; ++nt) {
      const int rbase = mBlk + wm * 32 + mt * 16 + hsel * 8;
      const int col   = nBlk + wn * 64 + nt * 16 + lm;
      #pragma unroll
      for (int i = 0; i < 8; ++i)
        Cm[(size_t)(rbase + i) * N + col] = acc[mt][nt][i];
    }
  }
}

// ---------------------------------------------------------------------------
// Bidirectional LSE-stabilized WKV scan. One lane per (b, dir, c) channel:
// 8192 channels = 256 wave32 waves; coalesced along C. Fuses sigmoid(r)*y
// and writes the result in place into the r slot of rkv (consumed by GEMM2).
// ---------------------------------------------------------------------------
__global__ __launch_bounds__(256)
void birwkv_scan(float* __restrict__ rkv,
                 const float* __restrict__ td,  const float* __restrict__ tf,
                 const float* __restrict__ tdr, const float* __restrict__ tfr,
                 int B, int T, int C) {
  const int tid = blockIdx.x * blockDim.x + threadIdx.x;
  const int c = tid % C;
  const int d = (tid / C) & 1;
  const int b = tid / (2 * C);
  if (b >= B) return;

  const float u = d ? tfr[c] : tf[c];
  const float w = -__expf(d ? tdr[c] : td[c]);
  const int row = 6 * C;
  float* base = rkv + (size_t)b * T * row + (size_t)d * 3 * C + c;

  float num = 0.0f, den = 0.0f, mx = -1e38f;
  const int t0 = d ? (T - 1) : 0;
  const int dt = d ? -1 : 1;

  for (int s = 0; s < T; ++s) {
    const int t = t0 + s * dt;
    float* p = base + (size_t)t * row;
    const float r = p[0];
    const float k = p[C];
    const float v = p[2 * C];

    const float ku = k + u;
    const float m  = fmaxf(mx, ku);
    const float e1 = __expf(mx - m);
    const float e2 = __expf(ku - m);
    const float y  = (e1 * num + e2 * v) / (e1 * den + e2);

    const float mw = mx + w;
    const float m2 = fmaxf(mw, k);
    const float es1 = __expf(mw - m2);
    const float es2 = __expf(k - m2);
    num = es1 * num + es2 * v;
    den = es1 * den + es2;
    mx  = m2;

    const float sr = 1.0f / (1.0f + __expf(-r));
    p[0] = sr * y;                      // overwrite r slot with sigmoid(r)*y
  }
}

// ---------------------------------------------------------------------------
extern "C" void kernel_launch(void* const* d_in, const int* in_sizes, int n_in,
                              void* d_out, int out_size, void* d_ws,
                              size_t ws_size, hipStream_t stream) {
  (void)in_sizes; (void)n_in; (void)out_size; (void)ws_size;
  const float* x    = (const float*)d_in[0];   // (B,T,C)
  const float* Wrkv = (const float*)d_in[1];   // (C, 6C)
  const float* Wout = (const float*)d_in[2];   // (2C, C)
  const float* td   = (const float*)d_in[3];
  const float* tf   = (const float*)d_in[4];
  const float* tdr  = (const float*)d_in[5];
  const float* tfr  = (const float*)d_in[6];
  float* out = (float*)d_out;

  const int B = 8, T = 4096, C = 512;
  const int M = B * T;                         // 32768 rows
  float* rkv = (float*)d_ws;                   // M * 6C fp32 = 384 MiB

  dim3 blk(NTHREADS);

  // GEMM1: rkv = x @ W_rkv   (M x 512 @ 512 x 3072)
  dim3 g1((6 * C) / BN, M / BM);
  wmma_gemm_f32<<<g1, blk, 0, stream>>>(x, Wrkv, rkv, M, 6 * C, C,
                                        /*lda=*/C, /*kSplit=*/1 << 30,
                                        /*kSkip=*/0);

  // Fused bidirectional WKV scan + sigmoid gate, in place into r slots.
  const int nscan = B * 2 * C;                 // 8192 channels
  birwkv_scan<<<nscan / 256, blk, 0, stream>>>(rkv, td, tf, tdr, tfr, B, T, C);

  // GEMM2: out = [out_f | out_b] @ W_out, reading A straight out of rkv:
  //   kk in [0,512)    -> cols 0..511    (d=0, r slot)
  //   kk in [512,1024) -> cols 1536..2047 (d=1, r slot) => +1024 skip
  dim3 g2(C / BN, M / BM);
  wmma_gemm_f32<<<g2, blk, 0, stream>>>(rkv, Wout, out, M, C, 2 * C,
                                        /*lda=*/6 * C, /*kSplit=*/C,
                                        /*kSkip=*/2 * C);
}